// HGNN_18511309045828
// MI455X (gfx1250) — compile-verified
//
#include <hip/hip_runtime.h>
#include <hip/hip_bf16.h>
#include <stdint.h>
#include <math.h>

// Problem constants (match the reference)
#define B_    4
#define N_    4096
#define CIN_  128
#define COUT_ 128
#define K_    10
#define L_    50
#define NNZ_  65536

typedef __attribute__((ext_vector_type(2))) float v2f;
typedef __attribute__((ext_vector_type(8))) float v8f;

// ---------------------------------------------------------------------------
// Dense GEMM: filt[m][n] = sum_k fea[m][k] * W[k][n]
// M = B*N = 16384, K = CIN = 128, N = COUT = 128.
// One wave (32 lanes) computes one 16x16 output tile via V_WMMA_F32_16X16X4_F32.
// A (16x4 f32): lanes 0-15 hold M=0..15, VGPR0 = K=kk, VGPR1 = K=kk+1 where
//               kk = 2*(lane>=16).   B (4x16): lanes hold N=0..15, same K split.
// D (16x16 f32): VGPR r -> M=r (lanes 0-15) / M=8+r (lanes 16-31), N = lane&15.
// ---------------------------------------------------------------------------
__global__ void __launch_bounds__(32) gemm_feaW(const float* __restrict__ fea,
                                                const float* __restrict__ W,
                                                float* __restrict__ filt) {
  const int lane = threadIdx.x;            // 0..31, wave32
  const int m0   = blockIdx.x * 16;        // row tile
  const int n0   = blockIdx.y * 16;        // col tile
  const int mrow = m0 + (lane & 15);
  const int ncol = n0 + (lane & 15);
  const int kk   = (lane >> 4) * 2;        // 0 or 2

  const float* __restrict__ arow = fea + (size_t)mrow * CIN_;

  v8f acc = {};
  #pragma unroll 4
  for (int k = 0; k < CIN_; k += 4) {
    v2f a, b;
    a.x = arow[k + kk];
    a.y = arow[k + kk + 1];
    b.x = W[(size_t)(k + kk)     * COUT_ + ncol];
    b.y = W[(size_t)(k + kk + 1) * COUT_ + ncol];
    acc = __builtin_amdgcn_wmma_f32_16x16x4_f32(
        /*neg_a=*/false, a, /*neg_b=*/false, b,
        /*c_mod=*/(short)0, acc, /*reuse_a=*/false, /*reuse_b=*/false);
  }

  const int drow = m0 + ((lane >> 4) ? 8 : 0);
  #pragma unroll
  for (int r = 0; r < 8; ++r)
    filt[(size_t)(drow + r) * COUT_ + ncol] = acc[r];
}

// ---------------------------------------------------------------------------
// Zero-fill
// ---------------------------------------------------------------------------
__global__ void fillz(float* __restrict__ p, unsigned n) {
  unsigned i = blockIdx.x * blockDim.x + threadIdx.x;
  if (i < n) p[i] = 0.0f;
}

// ---------------------------------------------------------------------------
// COO SpMM scatter: out[b, row, :] += v * (dscale?[col]) * X[b, col, :]
// One wave per nonzero; each lane owns 4 contiguous channels (float4 gather,
// 4x global_atomic_add_f32 scatter).
// ---------------------------------------------------------------------------
__global__ void __launch_bounds__(256) spmm_scatter(const int*   __restrict__ idx,
                                                    const float* __restrict__ val,
                                                    const float* __restrict__ X,
                                                    const float* __restrict__ dscale,
                                                    float* __restrict__ out) {
  const int b    = blockIdx.y;
  const int wave = threadIdx.x >> 5;
  const int lane = threadIdx.x & 31;
  const int e    = blockIdx.x * 8 + wave;            // 0..NNZ_-1 exactly

  const int r = idx[((size_t)b * 2)     * NNZ_ + e];
  const int c = idx[((size_t)b * 2 + 1) * NNZ_ + e];
  float v = val[(size_t)b * NNZ_ + e];
  if (dscale) v *= dscale[c];                        // fused diag(d) on gather side

  const float4 x = *(const float4*)(X + ((size_t)b * N_ + c) * COUT_ + lane * 4);
  float* o = out + ((size_t)b * N_ + r) * COUT_ + lane * 4;
  atomicAdd(o + 0, v * x.x);
  atomicAdd(o + 1, v * x.y);
  atomicAdd(o + 2, v * x.z);
  atomicAdd(o + 3, v * x.w);
}

// ---------------------------------------------------------------------------
// One RNN step: h_out[b,n] = tanh( x_b . W_ih[n] + h_b . W_hh[n] + b_ih[n] + b_hh[n] )
// x_b = item_emb[joblst[b, step]].  One wave per (b, n).
// h[b,:] (16 KB) is staged into LDS once per workgroup via the async
// global->LDS copy path (ASYNCcnt + s_wait_asynccnt + barrier), so the 8 waves
// of the block share one copy instead of re-reading it from cache. W_hh
// (64 MB, resident in the 192 MB L2) is streamed with float4 loads + prefetch.
// ---------------------------------------------------------------------------
__global__ void __launch_bounds__(256) rnn_step(const float*     __restrict__ item_emb,
                                                const long long* __restrict__ joblst,
                                                const float* __restrict__ W_ih,
                                                const float* __restrict__ W_hh,
                                                const float* __restrict__ b_ih,
                                                const float* __restrict__ b_hh,
                                                const float* __restrict__ h_in,
                                                float* __restrict__ h_out,
                                                int step) {
  __shared__ float hs[N_];                           // 16 KB
  const int b    = blockIdx.y;
  const int wave = threadIdx.x >> 5;
  const int lane = threadIdx.x & 31;
  const int n    = blockIdx.x * 8 + wave;            // 0..N_-1

  // ---- async stage h[b,:] -> LDS (each lane moves 16B x 4 iterations) ----
  {
    const float* hsrc = h_in + (size_t)b * N_;
    const unsigned lds_base = (unsigned)(uintptr_t)hs;   // LDS byte offset
    #pragma unroll
    for (int i = threadIdx.x * 4; i < N_; i += 256 * 4) {
      unsigned lds_addr = lds_base + (unsigned)i * 4u;
      unsigned long long ga = (unsigned long long)(uintptr_t)(hsrc + i);
      asm volatile("global_load_async_to_lds_b128 %0, %1, off"
                   :: "v"(lds_addr), "v"(ga) : "memory");
    }
    asm volatile("s_wait_asynccnt 0x0" ::: "memory");
    __syncthreads();
  }

  const long long ji = joblst[(size_t)b * L_ + step];
  const float* __restrict__ x   = item_emb + (size_t)ji * COUT_;
  const float* __restrict__ wih = W_ih + (size_t)n * COUT_;
  const float* __restrict__ whh = W_hh + (size_t)n * N_;

  float acc = 0.0f;
  {  // input projection: 128 elems = one float4 per lane
    float4 xa = *(const float4*)(x   + lane * 4);
    float4 wa = *(const float4*)(wih + lane * 4);
    acc += xa.x * wa.x + xa.y * wa.y + xa.z * wa.z + xa.w * wa.w;
  }
  for (int k = lane * 4; k < N_; k += 128) {         // hidden projection
    __builtin_prefetch(whh + k + 512, 0, 1);         // global_prefetch_b8
    float4 ha = *(const float4*)(hs  + k);           // ds_load_b128 from LDS
    float4 wa = *(const float4*)(whh + k);
    acc += ha.x * wa.x + ha.y * wa.y + ha.z * wa.z + ha.w * wa.w;
  }
  #pragma unroll
  for (int off = 16; off > 0; off >>= 1)
    acc += __shfl_xor(acc, off, 32);
  if (lane == 0)
    h_out[(size_t)b * N_ + n] = tanhf(acc + b_ih[n] + b_hh[n]);
}

// ---------------------------------------------------------------------------
// Head: out[b,n,k] = sigmoid( gelu(res[b,n,:] . dense_W[k,:] + dense_b[k])
//                             * gelu(Emb[b,n]) )       (exact erf-based gelu)
// ---------------------------------------------------------------------------
__global__ void __launch_bounds__(256) head_kernel(const float* __restrict__ res,
                                                   const float* __restrict__ dense_W,
                                                   const float* __restrict__ dense_b,
                                                   const float* __restrict__ Emb,
                                                   float* __restrict__ out) {
  const unsigned i = blockIdx.x * blockDim.x + threadIdx.x;
  const unsigned total = (unsigned)B_ * N_ * K_;
  if (i >= total) return;
  const int k = i % K_;
  const unsigned bn = i / K_;                        // b*N + n

  const float* __restrict__ rr = res + (size_t)bn * COUT_;
  const float* __restrict__ wk = dense_W + (size_t)k * COUT_;
  float acc = dense_b[k];
  #pragma unroll 8
  for (int c = 0; c < COUT_; c += 4) {
    float4 ra = *(const float4*)(rr + c);
    float4 wa = *(const float4*)(wk + c);
    acc += ra.x * wa.x + ra.y * wa.y + ra.z * wa.z + ra.w * wa.w;
  }
  const float inv_sqrt2 = 0.70710678118654752f;
  float g1 = 0.5f * acc * (1.0f + erff(acc * inv_sqrt2));
  float e  = Emb[bn];
  float g2 = 0.5f * e * (1.0f + erff(e * inv_sqrt2));
  float z  = g1 * g2;
  out[i] = 1.0f / (1.0f + expf(-z));
}

// ---------------------------------------------------------------------------
// Orchestration
// ---------------------------------------------------------------------------
extern "C" void kernel_launch(void* const* d_in, const int* in_sizes, int n_in,
                              void* d_out, int out_size, void* d_ws, size_t ws_size,
                              hipStream_t stream) {
  // setup_inputs() order:
  const int*       phi1_idx     = (const int*)d_in[0];
  const float*     phi1_val     = (const float*)d_in[1];
  const int*       phi1_inv_idx = (const int*)d_in[2];
  const float*     phi1_inv_val = (const float*)d_in[3];
  const int*       phi2_idx     = (const int*)d_in[4];
  const float*     phi2_val     = (const float*)d_in[5];
  const int*       phi2_inv_idx = (const int*)d_in[6];
  const float*     phi2_inv_val = (const float*)d_in[7];
  const float*     fea          = (const float*)d_in[8];
  const long long* joblst       = (const long long*)d_in[9];   // int64 in reference
  const float*     W1           = (const float*)d_in[10];
  const float*     d1           = (const float*)d_in[11];
  const float*     W2           = (const float*)d_in[12];
  const float*     d2           = (const float*)d_in[13];
  const float*     W_ih         = (const float*)d_in[14];
  const float*     W_hh         = (const float*)d_in[15];
  const float*     b_ih         = (const float*)d_in[16];
  const float*     b_hh         = (const float*)d_in[17];
  const float*     dense_W      = (const float*)d_in[18];
  const float*     dense_b      = (const float*)d_in[19];
  const float*     item_emb     = (const float*)d_in[20];
  float* out = (float*)d_out;

  // Workspace layout (floats)
  const size_t FEAT = (size_t)B_ * N_ * COUT_;   // 2,097,152
  const size_t HID  = (size_t)B_ * N_;           // 16,384
  float* ws   = (float*)d_ws;
  float* res  = ws;                              // [B,N,COUT] accumulator
  float* filt = ws + FEAT;                       // [B,N,COUT]
  float* tmp  = ws + 2 * FEAT;                   // [B,N,COUT]
  float* h0   = ws + 3 * FEAT;                   // [B,N]
  float* h1   = h0 + HID;                        // [B,N]

  const dim3 blk32(32), blk256(256);
  const dim3 gemm_grid((unsigned)((B_ * N_) / 16), COUT_ / 16);  // 1024 x 8
  const dim3 spmm_grid(NNZ_ / 8, B_);                            // 8192 x 4
  const dim3 rnn_grid(N_ / 8, B_);                               // 512 x 4

  // Zero accumulators
  fillz<<<(unsigned)(FEAT + 255) / 256, blk256, 0, stream>>>(res, (unsigned)FEAT);
  fillz<<<(unsigned)(HID + 255) / 256,  blk256, 0, stream>>>(h0,  (unsigned)HID);

  // --- Hypergraph layer 1: res += phi1 @ diag(d1) @ phi1_inv @ (fea @ W1)
  gemm_feaW<<<gemm_grid, blk32, 0, stream>>>(fea, W1, filt);
  fillz<<<(unsigned)(FEAT + 255) / 256, blk256, 0, stream>>>(tmp, (unsigned)FEAT);
  spmm_scatter<<<spmm_grid, blk256, 0, stream>>>(phi1_inv_idx, phi1_inv_val, filt, nullptr, tmp);
  spmm_scatter<<<spmm_grid, blk256, 0, stream>>>(phi1_idx, phi1_val, tmp, d1, res);

  // --- Hypergraph layer 2: res += phi2 @ diag(d2) @ phi2_inv @ (fea @ W2)
  gemm_feaW<<<gemm_grid, blk32, 0, stream>>>(fea, W2, filt);
  fillz<<<(unsigned)(FEAT + 255) / 256, blk256, 0, stream>>>(tmp, (unsigned)FEAT);
  spmm_scatter<<<spmm_grid, blk256, 0, stream>>>(phi2_inv_idx, phi2_inv_val, filt, nullptr, tmp);
  spmm_scatter<<<spmm_grid, blk256, 0, stream>>>(phi2_idx, phi2_val, tmp, d2, res);

  // --- RNN scan: 50 sequential steps, ping-pong h0/h1 (final lands in h0)
  for (int l = 0; l < L_; ++l) {
    const float* hin  = (l & 1) ? h1 : h0;
    float*       hout = (l & 1) ? h0 : h1;
    rnn_step<<<rnn_grid, blk256, 0, stream>>>(item_emb, joblst, W_ih, W_hh,
                                              b_ih, b_hh, hin, hout, l);
  }

  // --- Head: gelu(res @ dense_W^T + b) * gelu(Emb) -> sigmoid
  const unsigned total = (unsigned)B_ * N_ * K_;
  head_kernel<<<(total + 255) / 256, blk256, 0, stream>>>(res, dense_W, dense_b, h0, out);

  (void)in_sizes; (void)n_in; (void)out_size; (void)ws_size; (void)b_hh;
}